// HPWL_88441966559841
// MI455X (gfx1250) — compile-verified
//
#include <hip/hip_runtime.h>

// ---------------------------------------------------------------------------
// HPWL on MI455X (gfx1250, wave32).
// Memory-bound scatter-reduce: ~245 MB streaming reads (NT b128 loads) +
// 80 MB of L2-resident per-net min/max state (interleaved uint4 per net)
// updated with native no-return u32 atomics (GLOBAL_ATOMIC_MIN/MAX_U32).
// 192 MB L2 holds the entire net state; each pin's 4 atomics hit one 16B
// region. Final sum is a fixed-shape wave32 shuffle/LDS tree -> deterministic.
// ---------------------------------------------------------------------------

#define BLK 256

typedef float        v4f __attribute__((ext_vector_type(4)));
typedef int          v4i __attribute__((ext_vector_type(4)));
typedef unsigned int v4u __attribute__((ext_vector_type(4)));

// Monotone float -> u32 key (sign-magnitude to ordered): works for any float,
// lets us use single-instruction integer min/max atomics.
static __device__ __forceinline__ unsigned int f2ord(float f) {
    unsigned int u = __float_as_uint(f);
    return (u & 0x80000000u) ? ~u : (u | 0x80000000u);
}
static __device__ __forceinline__ float ord2f(unsigned int k) {
    unsigned int u = (k & 0x80000000u) ? (k & 0x7FFFFFFFu) : ~k;
    return __uint_as_float(u);
}

// Net state: uint4 per net = {xminKey, xmaxKey, yminKey, ymaxKey}.
// min-init = 0xFFFFFFFF (> every real key), max-init = 0 (< every real key).
// A net whose xminKey is still 0xFFFFFFFF after the scatter received no pins.
__global__ __launch_bounds__(BLK) void hpwl_init(v4u* __restrict__ state, int N) {
    int i = blockIdx.x * BLK + threadIdx.x;
    if (i < N) {
        v4u v = {0xFFFFFFFFu, 0u, 0xFFFFFFFFu, 0u};
        state[i] = v;                              // one global_store_b128
    }
}

static __device__ __forceinline__ void scatter_one(unsigned int* __restrict__ state,
                                                   float x, float y, int n) {
    unsigned int* base = state + 4u * (unsigned int)n;
    unsigned int kx = f2ord(x);
    unsigned int ky = f2ord(y);
    // Results unused -> no-return atomics (STOREcnt path); all 4 in one 16B region.
    atomicMin(base + 0, kx);
    atomicMax(base + 1, kx);
    atomicMin(base + 2, ky);
    atomicMax(base + 3, ky);
}

// Fast path: 4 pins per thread, b128 NT loads. Requires P % 4 == 0.
__global__ __launch_bounds__(BLK) void hpwl_scatter_v4(const float* __restrict__ pos,
                                                       const int*   __restrict__ p2n,
                                                       unsigned int* __restrict__ state,
                                                       int P) {
    int i4 = (blockIdx.x * BLK + threadIdx.x) * 4;
    if (i4 >= P) return;
    // Non-temporal: these 240 MB stream once; keep L2 for the net state.
    v4f x = __builtin_nontemporal_load((const v4f*)(pos + i4));
    v4f y = __builtin_nontemporal_load((const v4f*)(pos + P + i4));
    v4i n = __builtin_nontemporal_load((const v4i*)(p2n + i4));
    scatter_one(state, x.x, y.x, n.x);
    scatter_one(state, x.y, y.y, n.y);
    scatter_one(state, x.z, y.z, n.z);
    scatter_one(state, x.w, y.w, n.w);
}

// General path (any P / alignment).
__global__ __launch_bounds__(BLK) void hpwl_scatter_s(const float* __restrict__ pos,
                                                      const int*   __restrict__ p2n,
                                                      unsigned int* __restrict__ state,
                                                      int P) {
    int i = blockIdx.x * BLK + threadIdx.x;
    if (i >= P) return;
    float x = __builtin_nontemporal_load(pos + i);
    float y = __builtin_nontemporal_load(pos + P + i);
    int   n = __builtin_nontemporal_load(p2n + i);
    scatter_one(state, x, y, n);
}

// Wave32 + LDS deterministic block reduction.
static __device__ __forceinline__ float block_reduce(float v, float* sm) {
    #pragma unroll
    for (int d = 16; d > 0; d >>= 1)
        v += __shfl_down(v, d, 32);               // wave32 shuffle tree
    int lane = threadIdx.x & 31;
    int wid  = threadIdx.x >> 5;
    if (lane == 0) sm[wid] = v;
    __syncthreads();
    if (wid == 0) {
        v = (lane < (BLK >> 5)) ? sm[lane] : 0.0f;
        #pragma unroll
        for (int d = 4; d > 0; d >>= 1)           // BLK/32 = 8 partials
            v += __shfl_down(v, d, 32);
    }
    return v;                                      // valid in lane 0 of wave 0
}

__global__ __launch_bounds__(BLK) void hpwl_spans(const v4u* __restrict__ state,
                                                  const unsigned char* __restrict__ mask,
                                                  float* __restrict__ partials,
                                                  int N) {
    __shared__ float sm[BLK >> 5];
    int i = blockIdx.x * BLK + threadIdx.x;
    float s = 0.0f;
    if (i < N) {
        v4u st = __builtin_nontemporal_load(state + i);   // one b128, last use
        if (st.x != 0xFFFFFFFFu && mask[i]) {
            float sx = ord2f(st.y) - ord2f(st.x);
            float sy = ord2f(st.w) - ord2f(st.z);
            s = sx + sy;
        }
    }
    float t = block_reduce(s, sm);
    if (threadIdx.x == 0) partials[blockIdx.x] = t;
}

// Single block, fixed accumulation order -> bit-deterministic final sum.
__global__ __launch_bounds__(BLK) void hpwl_final(const float* __restrict__ partials,
                                                  int nparts,
                                                  float* __restrict__ out) {
    __shared__ float sm[BLK >> 5];
    float s = 0.0f;
    for (int i = threadIdx.x; i < nparts; i += BLK)
        s += partials[i];
    float t = block_reduce(s, sm);
    if (threadIdx.x == 0) out[0] = t;
}

extern "C" void kernel_launch(void* const* d_in, const int* in_sizes, int n_in,
                              void* d_out, int out_size, void* d_ws, size_t ws_size,
                              hipStream_t stream) {
    const float*         pos  = (const float*)d_in[0];
    const int*           p2n  = (const int*)d_in[1];
    const unsigned char* mask = (const unsigned char*)d_in[2];

    const int P = in_sizes[1];   // NUM_PINS  (pos has 2*P elements)
    const int N = in_sizes[2];   // NUM_NETS

    // Workspace layout: N x uint4 net state, then block partials.
    v4u*   state    = (v4u*)d_ws;
    float* partials = (float*)((unsigned int*)d_ws + 4u * (size_t)N);

    const int gInit = (N + BLK - 1) / BLK;
    const int gSpan = (N + BLK - 1) / BLK;

    hpwl_init<<<gInit, BLK, 0, stream>>>(state, N);

    if ((P & 3) == 0) {
        const int gScat = (P / 4 + BLK - 1) / BLK;
        hpwl_scatter_v4<<<gScat, BLK, 0, stream>>>(pos, p2n, (unsigned int*)state, P);
    } else {
        const int gScat = (P + BLK - 1) / BLK;
        hpwl_scatter_s<<<gScat, BLK, 0, stream>>>(pos, p2n, (unsigned int*)state, P);
    }

    hpwl_spans<<<gSpan, BLK, 0, stream>>>(state, mask, partials, N);
    hpwl_final<<<1,     BLK, 0, stream>>>(partials, gSpan, (float*)d_out);
}